// MaskRCNN_65025804861596
// MI455X (gfx1250) — compile-verified
//
#include <hip/hip_runtime.h>
#include <stdint.h>

// Problem constants (from reference)
#define NN 2
#define MM 128
#define RR 8
#define HH 224
#define WW 224
#define CC 81
#define TRAIN_ROIS 1024
#define NNEG 768
#define NPOS 256

// Mask streaming config
#define CHUNK_ROWS 32
#define CHUNK_ELTS (CHUNK_ROWS * WW)        // 7168 floats = 28672 B per chunk
#define NCHUNK (HH / CHUNK_ROWS)            // 7 chunks per 224x224 slice
#define F4_PER_CHUNK (CHUNK_ELTS / 4)       // 1792
#define MASK_THREADS 256
#define F4_PER_THREAD (F4_PER_CHUNK / MASK_THREADS) // 7

typedef float    vfloat4 __attribute__((ext_vector_type(4)));
typedef uint32_t uint32x4 __attribute__((ext_vector_type(4)));
typedef int32_t  int32x8  __attribute__((ext_vector_type(8)));
typedef int32_t  int32x4  __attribute__((ext_vector_type(4)));

// --- pos flag, replicated bug-for-bug from the reference _pairwise_iou ---
__device__ __forceinline__ float pos_flag(float px1, float py1, float px2, float py2,
                                          float gx1, float gy1, float gx2, float gy2) {
  float a1 = (px2 - px1 + 1.f) * (py2 - py1);       // height has no +1 (reference)
  float a2 = (gx2 - gx1 + 1.f) * (gy2 - gy1);
  float ix1 = fmaxf(px1, gy1);                       // reference uses g[...,1] here (bug kept)
  float iy1 = fmaxf(py1, gy1);
  float ix2 = fminf(px2, gx2);
  float iy2 = fminf(py2, gy2);
  float iw = fmaxf(0.f, ix2 - ix1 + 1.f);
  float ih = fmaxf(0.f, iy2 - iy1 + 1.f);
  float inter = iw * ih;
  float iou = inter / (a1 + a2 + inter);             // denominator per reference
  return (iou >= 0.5f) ? 1.f : 0.f;
}

// ---------------------------------------------------------------------------
// Kernel A: rois / cls_t / bbox_t  (tiny: 2048 slots, fully cache-resident)
// ---------------------------------------------------------------------------
__global__ __launch_bounds__(128)
void targets_kernel(const float* __restrict__ proposals,
                    const float* __restrict__ gt_classes,
                    const float* __restrict__ gt_bboxes,
                    const int*   __restrict__ idx_neg,
                    const int*   __restrict__ idx_pos,
                    float* __restrict__ rois,
                    float* __restrict__ cls_t,
                    float* __restrict__ bbox_t) {
  int slot = blockIdx.x;            // n*1024 + j
  int n = slot >> 10;
  int j = slot & 1023;
  bool is_neg = (j < NNEG);
  int idx = is_neg ? idx_neg[j] : idx_pos[j - NNEG];
  int m = idx / RR;
  int r = idx - m * RR;

  const float* p = proposals + (size_t)(n * MM + m) * 4;
  const float* g = gt_bboxes + (size_t)(n * RR + r) * 4;
  float px1 = p[0], py1 = p[1], px2 = p[2], py2 = p[3];
  float gx1 = g[0], gy1 = g[1], gx2 = g[2], gy2 = g[3];

  float pos = pos_flag(px1, py1, px2, py2, gx1, gy1, gx2, gy2);
  float f = is_neg ? (1.f - pos) : pos;

  int t = threadIdx.x;
  if (t < 4) {
    float pv[4] = {px1, py1, px2, py2};
    rois[(size_t)slot * 4 + t] = pv[t] * f;
    float bt;
    if (is_neg) {
      // ctr = [floor(w/2)+x1, floor(h/2)+y1, w, h]
      float w = gx2 - gx1 + 1.f;
      float h = gy2 - gy1 + 1.f;
      float cv[4] = {floorf(w * 0.5f) + gx1, floorf(h * 0.5f) + gy1, w, h};
      bt = cv[t];
    } else {
      float gv[4] = {gx1, gy1, gx2, gy2};
      bt = gv[t];
    }
    bbox_t[(size_t)slot * 4 + t] = bt * f;
  }
  const float* cls_src = gt_classes + (size_t)(n * RR + r) * CC;
  for (int c = t; c < CC; c += blockDim.x)
    cls_t[(size_t)slot * CC + c] = cls_src[c] * f;
}

// ---------------------------------------------------------------------------
// Kernel B: mask_t — TDM (tensor_load_to_lds) double-buffered stream
// ---------------------------------------------------------------------------
// Issue one TDM descriptor: 1-D tile of CHUNK_ELTS f32 elements, global -> LDS.
// D# layout per CDNA5 ISA §8 (group0: count/lds_addr/global_addr/type,
// group1: data_size + dims/strides; remaining groups unused -> zeros).
// This toolchain's builtin is the 6-arg form:
//   (uint32x4 g0, int32x8 g1, int32x4, int32x4, int32x8, i32 cpol)
__device__ __forceinline__ void tdm_issue_chunk(uint64_t gaddr, uint32_t lds_addr) {
  uint32x4 g0;
  g0[0] = 1u;                                                  // count=1, user mode
  g0[1] = lds_addr;                                            // LDS byte address
  g0[2] = (uint32_t)(gaddr & 0xFFFFFFFFu);                     // global_addr[31:0]
  g0[3] = (uint32_t)((gaddr >> 32) & 0x1FFFFFFu) | (2u << 30); // addr[56:32] | type=2
  int32x8 g1;
  g1[0] = (2 << 16);                       // workgroup_mask=0, data_size=2 (4B)
  g1[1] = (int32_t)(CHUNK_ELTS << 16);     // tensor_dim0[15:0] in bits 63:48
  g1[2] = (1 << 16);                       // tensor_dim0 hi=0 | tensor_dim1=1
  g1[3] = (int32_t)(CHUNK_ELTS << 16);     // tile_dim0 = CHUNK_ELTS
  g1[4] = 1;                               // tile_dim1 = 1, tile_dim2 = 0
  g1[5] = (int32_t)CHUNK_ELTS;             // tensor_dim0_stride low
  g1[6] = 0;
  g1[7] = 0;
  int32x4 z4 = {0, 0, 0, 0};
  int32x8 z8 = {0, 0, 0, 0, 0, 0, 0, 0};
  __builtin_amdgcn_tensor_load_to_lds(g0, g1, z4, z4, z8, 0);
}

__global__ __launch_bounds__(MASK_THREADS)
void mask_kernel(const float* __restrict__ proposals,
                 const float* __restrict__ gt_bboxes,
                 const float* __restrict__ gt_masks,
                 const int*   __restrict__ idx_pos,
                 float* __restrict__ mask_out) {
  __shared__ __align__(16) float lds[2][CHUNK_ELTS];

  int blk = blockIdx.x;             // n*NPOS + j
  int n = blk / NPOS;
  int j = blk - n * NPOS;
  int idx = idx_pos[j];
  int m = idx / RR;
  int k = idx - m * RR;

  const float* p = proposals + (size_t)(n * MM + m) * 4;
  const float* g = gt_bboxes + (size_t)(n * RR + k) * 4;
  float px1 = p[0], py1 = p[1], px2 = p[2], py2 = p[3];
  float pos = pos_flag(px1, py1, px2, py2, g[0], g[1], g[2], g[3]);

  float* out = mask_out + (size_t)blk * (HH * WW);
  int t = threadIdx.x;

  if (pos == 0.f) {
    // whole slice is zero; skip the 200KB gt_masks read entirely
    vfloat4 z = {0.f, 0.f, 0.f, 0.f};
    vfloat4* o4 = (vfloat4*)out;
    for (int i = t; i < (HH * WW) / 4; i += MASK_THREADS)
      __builtin_nontemporal_store(z, o4 + i);
    return;
  }

  // pos == 1: rp == proposal coords. keep_r uses rp[0]/rp[2] vs ROW,
  // keep_c uses rp[1]/rp[3] vs COL (reference's swapped convention kept).
  const float* src = gt_masks + (size_t)(n * RR + k) * (HH * WW);
  uint64_t gbase = (uint64_t)(uintptr_t)src;
  // low 32 bits of a generic shared pointer == LDS byte offset (ISA §10.2 aperture)
  uint32_t lds_a = (uint32_t)(uintptr_t)&lds[0][0];
  uint32_t lds_b = (uint32_t)(uintptr_t)&lds[1][0];
  bool issuer = (t < 32);           // wave 0 only (other waves branch around via EXECZ)

  if (issuer) tdm_issue_chunk(gbase, lds_a);

  for (int c = 0; c < NCHUNK; ++c) {
    if (issuer) {
      if (c + 1 < NCHUNK) {
        // prefetch next chunk into the other buffer (consumed-before barrier below)
        tdm_issue_chunk(gbase + (uint64_t)(c + 1) * CHUNK_ELTS * 4,
                        ((c + 1) & 1) ? lds_b : lds_a);
        __builtin_amdgcn_s_wait_tensorcnt(1);   // chunk c complete (in-order TDM)
      } else {
        __builtin_amdgcn_s_wait_tensorcnt(0);
      }
    }
    __syncthreads();                 // chunk c data visible to all waves

    const vfloat4* bp = (const vfloat4*)&lds[c & 1][0];
    vfloat4* o4 = (vfloat4*)(out + (size_t)c * CHUNK_ELTS);
#pragma unroll
    for (int i = 0; i < F4_PER_THREAD; ++i) {
      int f4 = t + i * MASK_THREADS;
      vfloat4 v = bp[f4];
      int rowInChunk = f4 / (WW / 4);               // 56 float4 per row
      int colf4 = f4 - rowInChunk * (WW / 4);
      float rowf = (float)(c * CHUNK_ROWS + rowInChunk);
      float keep_r = (rowf >= px1 && rowf < px2) ? 1.f : 0.f;
      float c0 = (float)(colf4 * 4);
      vfloat4 res;
      res.x = v.x * keep_r * (((c0 + 0.f) >= py1 && (c0 + 0.f) < py2) ? 1.f : 0.f);
      res.y = v.y * keep_r * (((c0 + 1.f) >= py1 && (c0 + 1.f) < py2) ? 1.f : 0.f);
      res.z = v.z * keep_r * (((c0 + 2.f) >= py1 && (c0 + 2.f) < py2) ? 1.f : 0.f);
      res.w = v.w * keep_r * (((c0 + 3.f) >= py1 && (c0 + 3.f) < py2) ? 1.f : 0.f);
      __builtin_nontemporal_store(res, o4 + f4);
    }
    __syncthreads();                 // reads done before buffer is DMA-overwritten
  }
}

// ---------------------------------------------------------------------------
extern "C" void kernel_launch(void* const* d_in, const int* in_sizes, int n_in,
                              void* d_out, int out_size, void* d_ws, size_t ws_size,
                              hipStream_t stream) {
  (void)in_sizes; (void)n_in; (void)out_size; (void)d_ws; (void)ws_size;
  const float* proposals  = (const float*)d_in[0];
  const float* gt_classes = (const float*)d_in[1];
  const float* gt_bboxes  = (const float*)d_in[2];
  const float* gt_masks   = (const float*)d_in[3];
  const int*   idx_neg    = (const int*)d_in[4];
  const int*   idx_pos    = (const int*)d_in[5];

  float* out    = (float*)d_out;
  float* rois   = out;                                   // 2*1024*4
  float* cls_t  = rois  + (size_t)NN * TRAIN_ROIS * 4;   // 2*1024*81
  float* bbox_t = cls_t + (size_t)NN * TRAIN_ROIS * CC;  // 2*1024*4
  float* mask_t = bbox_t + (size_t)NN * TRAIN_ROIS * 4;  // 2*256*224*224

  targets_kernel<<<NN * TRAIN_ROIS, 128, 0, stream>>>(
      proposals, gt_classes, gt_bboxes, idx_neg, idx_pos, rois, cls_t, bbox_t);

  mask_kernel<<<NN * NPOS, MASK_THREADS, 0, stream>>>(
      proposals, gt_bboxes, gt_masks, idx_pos, mask_t);
}